// GCN_32169305047308
// MI455X (gfx1250) — compile-verified
//
#include <hip/hip_runtime.h>

typedef float v2f __attribute__((ext_vector_type(2)));
typedef float v8f __attribute__((ext_vector_type(8)));

#define VSIZE 512
#define F1 8
#define F2 3

// ---------------- degree / normalization ----------------

__global__ void k_deg_init(float* __restrict__ deg, int n) {
    int i = blockIdx.x * blockDim.x + threadIdx.x;
    if (i < n) deg[i] = 1.0f;  // self loop contributes 1 to every node's degree
}

__global__ void k_deg_edges(const long long* __restrict__ col,
                            float* __restrict__ deg, int e) {
    int i = blockIdx.x * blockDim.x + threadIdx.x;
    if (i < e) atomicAdd(&deg[(int)col[i]], 1.0f);
}

__global__ void k_dinv(const float* __restrict__ deg,
                       float* __restrict__ dinv, int n) {
    int i = blockIdx.x * blockDim.x + threadIdx.x;
    if (i < n) dinv[i] = rsqrtf(deg[i]);  // deg >= 1 always (self loops)
}

// ---------------- layer 1 GEMM: h1[N,8] = x[N,512] @ W1[512,8] (fp32 WMMA) --

__global__ void __launch_bounds__(256)
k_gemm1(const float* __restrict__ x, const float* __restrict__ W1,
        float* __restrict__ h1, int n) {
    // W1 padded to 16 columns, stored as K-pairs: Wp[p*16 + c] = {W(2p,c), W(2p+1,c)}
    __shared__ v2f Wp[(VSIZE / 2) * 16];
    for (int idx = threadIdx.x; idx < (VSIZE / 2) * 16; idx += blockDim.x) {
        int p = idx >> 4, c = idx & 15;
        v2f w;
        w.x = (c < F1) ? W1[(2 * p + 0) * F1 + c] : 0.0f;
        w.y = (c < F1) ? W1[(2 * p + 1) * F1 + c] : 0.0f;
        Wp[idx] = w;
    }
    __syncthreads();

    const int wave = threadIdx.x >> 5;          // 8 waves / block
    const int lane = threadIdx.x & 31;
    const int g    = lane >> 4;                 // lane group (0: lanes 0-15, 1: 16-31)
    const int m    = lane & 15;                 // A row within tile / B,C column index
    const int tile = blockIdx.x * 8 + wave;     // 16-row tile of x
    const int row0 = tile * 16;
    if (row0 >= n) return;                      // wave-uniform: EXEC stays all-ones

    // A operand: lane holds x[row0+m][k + 2g + {0,1}]  (ISA 16x4 f32 A layout)
    int arow = row0 + m;
    if (arow >= n) arow = n - 1;                // tail safety (N%16==0 normally)
    const v2f* __restrict__ xa = (const v2f*)(x + (size_t)arow * VSIZE + 2 * g);

    v8f acc = {};
    #pragma unroll 4
    for (int k = 0; k < VSIZE; k += 4) {
        v2f a = xa[k >> 1];                               // K = k+2g, k+2g+1
        v2f b = Wp[((k + 2 * g) >> 1) * 16 + m];          // B rows K=k+2g, k+2g+1, col m
        acc = __builtin_amdgcn_wmma_f32_16x16x4_f32(
            /*neg_a=*/false, a, /*neg_b=*/false, b,
            /*c_mod=*/(short)0, acc, /*reuse_a=*/false, /*reuse_b=*/false);
    }

    // C/D layout: VGPR r -> M = r + 8*g, N = lane&15; keep only N < 8
    if (row0 + 16 <= n) {
        // full tile (always the case for N % 16 == 0): branch-free epilogue
        if (m < F1) {
            float* __restrict__ hp = h1 + (size_t)(row0 + 8 * g) * F1 + m;
            #pragma unroll
            for (int r = 0; r < 8; ++r) hp[(size_t)r * F1] = acc[r];
        }
    } else {
        // ragged tail (not hit for this workload)
        if (m < F1) {
            #pragma unroll
            for (int r = 0; r < 8; ++r) {
                int orow = row0 + r + 8 * g;
                if (orow < n) h1[(size_t)orow * F1 + m] = acc[r];
            }
        }
    }
}

// ---------------- layer 1 aggregation ----------------

__global__ void k_agg1_init(const float* __restrict__ h1,
                            const float* __restrict__ dinv,
                            const float* __restrict__ b1,
                            float* __restrict__ out1, int n8) {
    int i = blockIdx.x * blockDim.x + threadIdx.x;
    if (i < n8) {
        int node = i >> 3, f = i & 7;
        float d = dinv[node];
        out1[i] = b1[f] + h1[i] * d * d;  // bias + self-loop message
    }
}

__global__ void k_agg1_edges(const long long* __restrict__ ei,
                             const float* __restrict__ h1,
                             const float* __restrict__ dinv,
                             float* __restrict__ out1, int e) {
    int i = blockIdx.x * blockDim.x + threadIdx.x;
    if (i < e) {
        int r = (int)ei[i];              // source
        int c = (int)ei[(size_t)e + i];  // target
        float nrm = dinv[r] * dinv[c];
        const float* hr = h1 + (size_t)r * F1;
        float* oc = out1 + (size_t)c * F1;
        #pragma unroll
        for (int f = 0; f < F1; ++f) atomicAdd(&oc[f], hr[f] * nrm);
    }
}

// ---------------- layer 2 local: z = relu(out1) @ W2; out2 init ------------

__global__ void k_layer2(const float* __restrict__ out1,
                         const float* __restrict__ dinv,
                         const float* __restrict__ W2,
                         const float* __restrict__ b2,
                         float* __restrict__ z,
                         float* __restrict__ out2, int n) {
    int i = blockIdx.x * blockDim.x + threadIdx.x;
    if (i < n) {
        float h[F1];
        #pragma unroll
        for (int f = 0; f < F1; ++f) h[f] = fmaxf(out1[(size_t)i * F1 + f], 0.0f);
        float d = dinv[i];
        #pragma unroll
        for (int c = 0; c < F2; ++c) {
            float s = 0.0f;
            #pragma unroll
            for (int f = 0; f < F1; ++f) s += h[f] * W2[f * F2 + c];
            z[(size_t)i * F2 + c]    = s;
            out2[(size_t)i * F2 + c] = b2[c] + s * d * d;  // bias + self loop
        }
    }
}

__global__ void k_agg2_edges(const long long* __restrict__ ei,
                             const float* __restrict__ z,
                             const float* __restrict__ dinv,
                             float* __restrict__ out2, int e) {
    int i = blockIdx.x * blockDim.x + threadIdx.x;
    if (i < e) {
        int r = (int)ei[i];
        int c = (int)ei[(size_t)e + i];
        float nrm = dinv[r] * dinv[c];
        const float* zr = z + (size_t)r * F2;
        float* oc = out2 + (size_t)c * F2;
        #pragma unroll
        for (int f = 0; f < F2; ++f) atomicAdd(&oc[f], zr[f] * nrm);
    }
}

// ---------------- log softmax over 3 classes ----------------

__global__ void k_lsm(const float* __restrict__ out2,
                      float* __restrict__ out, int n) {
    int i = blockIdx.x * blockDim.x + threadIdx.x;
    if (i < n) {
        float a = out2[(size_t)i * F2 + 0];
        float b = out2[(size_t)i * F2 + 1];
        float c = out2[(size_t)i * F2 + 2];
        float m = fmaxf(a, fmaxf(b, c));
        float lse = m + logf(expf(a - m) + expf(b - m) + expf(c - m));
        out[(size_t)i * F2 + 0] = a - lse;
        out[(size_t)i * F2 + 1] = b - lse;
        out[(size_t)i * F2 + 2] = c - lse;
    }
}

// ---------------- launch ----------------

extern "C" void kernel_launch(void* const* d_in, const int* in_sizes, int n_in,
                              void* d_out, int out_size, void* d_ws, size_t ws_size,
                              hipStream_t stream) {
    const float*     x  = (const float*)d_in[0];
    const long long* ei = (const long long*)d_in[1];  // int64 per reference
    const float*     W1 = (const float*)d_in[2];
    const float*     b1 = (const float*)d_in[3];
    const float*     W2 = (const float*)d_in[4];
    const float*     b2 = (const float*)d_in[5];
    float*           out = (float*)d_out;

    const int n = in_sizes[0] / VSIZE;   // 200000
    const int e = in_sizes[1] / 2;       // 6400000

    // carve workspace (all regions fully rewritten every call)
    char* w = (char*)d_ws;
    auto alignup = [](size_t v) { return (v + 255) & ~(size_t)255; };
    float* deg  = (float*)w; w += alignup((size_t)n * 4);
    float* dinv = (float*)w; w += alignup((size_t)n * 4);
    float* h1   = (float*)w; w += alignup((size_t)n * F1 * 4);
    float* out1 = (float*)w; w += alignup((size_t)n * F1 * 4);
    float* z    = (float*)w; w += alignup((size_t)n * F2 * 4);
    float* out2 = (float*)w; w += alignup((size_t)n * F2 * 4);

    const int B = 256;
    k_deg_init <<<(n + B - 1) / B, B, 0, stream>>>(deg, n);
    k_deg_edges<<<(e + B - 1) / B, B, 0, stream>>>(ei + e, deg, e);
    k_dinv     <<<(n + B - 1) / B, B, 0, stream>>>(deg, dinv, n);

    const int tiles   = (n + 15) / 16;        // 12500 (exact)
    const int gblocks = (tiles + 7) / 8;      // 8 waves per block
    k_gemm1<<<gblocks, B, 0, stream>>>(x, W1, h1, n);

    k_agg1_init <<<((n * F1) + B - 1) / B, B, 0, stream>>>(h1, dinv, b1, out1, n * F1);
    k_agg1_edges<<<(e + B - 1) / B, B, 0, stream>>>(ei, h1, dinv, out1, e);
    k_layer2    <<<(n + B - 1) / B, B, 0, stream>>>(out1, dinv, W2, b2, z, out2, n);
    k_agg2_edges<<<(e + B - 1) / B, B, 0, stream>>>(ei, z, dinv, out2, e);
    k_lsm       <<<(n + B - 1) / B, B, 0, stream>>>(out2, out, n);
}